// KANDecoder_13365938225274
// MI455X (gfx1250) — compile-verified
//
#include <hip/hip_runtime.h>
#include <hip/hip_bf16.h>
#include <math.h>

// ---------------------------------------------------------------------------
// KAN decoder, MI455X (gfx1250, wave32).
//   h0 = mean(x, axis=1)            : 1.07 GB read -> HBM-bound, ~46us floor
//   h1 = kan_layer0(h0)             : fused (1024 x 4608) @ (4608 x 64) GEMM
//                                     via V_WMMA_F32_16X16X4_F32
//   out = softplus(kan2(kan1(h1)))  : scalar tail (tiny)
// ---------------------------------------------------------------------------

typedef __attribute__((ext_vector_type(2))) float v2f;
typedef __attribute__((ext_vector_type(8))) float v8f;

#define N_IN    512      // d_in of layer 0
#define N_HID   64       // hidden
#define N_OUT   4        // d_out
#define NG      8        // G + K basis functions kept
#define NFEAT   9        // silu + 8 spline basis
#define K_TOT   (N_IN * NFEAT)     // 4608
#define CHUNK_I 16                 // inputs per LDS chunk
#define CHUNK_K (CHUNK_I * NFEAT)  // 144
#define A_STRIDE 148               // LDS row stride (bank-conflict pad)

__device__ __forceinline__ float silu_f(float x) {
    return x / (1.0f + expf(-x));
}

// Cox-de Boor, uniform grid: knots p[j] = (j-3)*0.4 - 1, j = 0..11 -> 8 basis
__device__ __forceinline__ void bspl8(float x, float* __restrict__ B) {
    const float h = 0.4f;
    float Bc[11];
#pragma unroll
    for (int j = 0; j < 11; ++j) {
        float p0 = (float)(j - 3) * h - 1.0f;
        float p1 = (float)(j - 2) * h - 1.0f;
        Bc[j] = (x >= p0 && x < p1) ? 1.0f : 0.0f;
    }
#pragma unroll
    for (int d = 1; d <= 3; ++d) {
        float inv = 1.0f / ((float)d * h);
#pragma unroll
        for (int m = 0; m + d < 11; ++m) {
            float pm   = (float)(m - 3) * h - 1.0f;      // p[m]
            float pmd1 = (float)(m + d - 2) * h - 1.0f;  // p[m+d+1]
            Bc[m] = (x - pm) * inv * Bc[m] + (pmd1 - x) * inv * Bc[m + 1];
        }
    }
#pragma unroll
    for (int g = 0; g < NG; ++g) B[g] = Bc[g];
}

// ---------------------------------------------------------------------------
// Kernel 1: h0[n][d] = mean_r x[n][r][d].  One block per n, 128 thr x float4
// covers the whole 512-float row; loads are fully coalesced 2KB lines.
// ---------------------------------------------------------------------------
__global__ void mean_rows_kernel(const float* __restrict__ x,
                                 float* __restrict__ h0) {
    const int n  = blockIdx.x;
    const int d4 = threadIdx.x;            // 0..127
    const float4* xp = reinterpret_cast<const float4*>(x)
                     + (size_t)n * (512u * 128u) + d4;
    float4 acc = make_float4(0.f, 0.f, 0.f, 0.f);
#pragma unroll 4
    for (int r = 0; r < 512; ++r) {
        float4 v = xp[(size_t)r * 128u];
        acc.x += v.x; acc.y += v.y; acc.z += v.z; acc.w += v.w;
    }
    const float inv = 1.0f / 512.0f;
    float4 o = make_float4(acc.x * inv, acc.y * inv, acc.z * inv, acc.w * inv);
    reinterpret_cast<float4*>(h0)[(size_t)n * 128u + d4] = o;
}

// ---------------------------------------------------------------------------
// Kernel 2: pack fused layer-0 weights  W0[k][o], k = i*9 + f:
//   f==0 -> sb0[o][i];  f>=1 -> coef0[o][i][f-1] * ss0[o][i]
// ---------------------------------------------------------------------------
__global__ void pack_w0_kernel(const float* __restrict__ coef0,
                               const float* __restrict__ sb0,
                               const float* __restrict__ ss0,
                               float* __restrict__ W0) {
    int idx = blockIdx.x * blockDim.x + threadIdx.x;
    if (idx >= K_TOT * N_HID) return;
    int k = idx / N_HID;
    int o = idx - k * N_HID;
    int i = k / NFEAT;
    int f = k - i * NFEAT;
    float w;
    if (f == 0) {
        w = sb0[o * N_IN + i];
    } else {
        w = coef0[(o * N_IN + i) * NG + (f - 1)] * ss0[o * N_IN + i];
    }
    W0[idx] = w;
}

// ---------------------------------------------------------------------------
// Kernel 3: layer-0 GEMM, h1 = F(h0) @ W0, F built on the fly into LDS.
// One block = 16 rows of the batch, 4 waves, wave w owns N-cols [16w,16w+16).
// K streamed in chunks of 16 inputs (144 K values) -> 36 WMMAs/chunk/wave.
// ---------------------------------------------------------------------------
__global__ void
__launch_bounds__(128)
layer0_wmma_kernel(const float* __restrict__ h0,
                   const float* __restrict__ W0,
                   float* __restrict__ h1) {
    __shared__ float Atile[16][A_STRIDE];

    const int rowBase = blockIdx.x * 16;
    const int lane = threadIdx.x & 31;
    const int wv   = threadIdx.x >> 5;       // 0..3 -> N tile
    const int mrow = lane & 15;              // M row / N col for frags
    const int hi   = lane >> 4;              // K half-select

    v8f acc = {};

    for (int c = 0; c < N_IN / CHUNK_I; ++c) {
        const int i0 = c * CHUNK_I;
        __syncthreads();   // previous chunk's WMMA reads done
        // --- build feature tile: 16 rows x 16 inputs x 9 features ---
        for (int p = threadIdx.x; p < 16 * CHUNK_I; p += 128) {
            int r  = p >> 4;
            int ii = p & 15;
            float xv = h0[(size_t)(rowBase + r) * N_IN + (i0 + ii)];
            float f[NFEAT];
            f[0] = silu_f(xv);
            bspl8(xv, f + 1);
#pragma unroll
            for (int q = 0; q < NFEAT; ++q)
                Atile[r][ii * NFEAT + q] = f[q];
        }
        __syncthreads();   // tile visible to all waves

        // --- WMMA over this chunk's 144 K values ---
        const int kbase = i0 * NFEAT;
#pragma unroll 4
        for (int kk = 0; kk < CHUNK_K; kk += 4) {
            v2f a, b;
            // A 16x4 f32 frag: lanes 0-15 hold K {kk,kk+1}, 16-31 {kk+2,kk+3}
            a.x = Atile[mrow][kk + 2 * hi + 0];
            a.y = Atile[mrow][kk + 2 * hi + 1];
            // B 4x16 f32 frag: col = lane&15, same K split
            const float* wp = W0 + (size_t)(kbase + kk + 2 * hi) * N_HID
                                 + (wv * 16 + mrow);
            b.x = wp[0];
            b.y = wp[N_HID];
            acc = __builtin_amdgcn_wmma_f32_16x16x4_f32(
                false, a, false, b, (short)0, acc, false, false);
        }
    }

    // C/D layout: VGPR v -> M = v + 8*hi, N = lane&15
#pragma unroll
    for (int v = 0; v < 8; ++v) {
        int m = v + 8 * hi;
        int ncol = wv * 16 + mrow;
        h1[(size_t)(rowBase + m) * N_HID + ncol] = acc[v];
    }
}

// ---------------------------------------------------------------------------
// Kernel 4: tail — layer1 (64->4), layer2 (4->4), softplus, scatter.
// One thread per batch row; ~3K MACs/thread, negligible.
// ---------------------------------------------------------------------------
__global__ void tail_kernel(const float* __restrict__ h1,
                            const float* __restrict__ coef1,
                            const float* __restrict__ sb1,
                            const float* __restrict__ ss1,
                            const float* __restrict__ coef2,
                            const float* __restrict__ sb2,
                            const float* __restrict__ ss2,
                            float* __restrict__ out, int nrows) {
    int n = blockIdx.x * blockDim.x + threadIdx.x;
    if (n >= nrows) return;

    const float* hr = h1 + (size_t)n * N_HID;
    float a1[N_OUT] = {0.f, 0.f, 0.f, 0.f};
    for (int i = 0; i < N_HID; ++i) {
        float xv = hr[i];
        float s = silu_f(xv);
        float B[NG];
        bspl8(xv, B);
#pragma unroll
        for (int o = 0; o < N_OUT; ++o) {
            const float* cp = coef1 + (size_t)(o * N_HID + i) * NG;
            float sp = 0.f;
#pragma unroll
            for (int g = 0; g < NG; ++g) sp += B[g] * cp[g];
            a1[o] += s * sb1[o * N_HID + i] + sp * ss1[o * N_HID + i];
        }
    }

    float a2[N_OUT] = {0.f, 0.f, 0.f, 0.f};
#pragma unroll
    for (int i = 0; i < N_OUT; ++i) {
        float xv = a1[i];
        float s = silu_f(xv);
        float B[NG];
        bspl8(xv, B);
#pragma unroll
        for (int o = 0; o < N_OUT; ++o) {
            const float* cp = coef2 + (size_t)(o * N_OUT + i) * NG;
            float sp = 0.f;
#pragma unroll
            for (int g = 0; g < NG; ++g) sp += B[g] * cp[g];
            a2[o] += s * sb2[o * N_OUT + i] + sp * ss2[o * N_OUT + i];
        }
    }

#pragma unroll
    for (int o = 0; o < N_OUT; ++o) {
        float v = a2[o];
        float p = (v > 20.0f) ? v : log1pf(expf(v));  // softplus
        out[(size_t)o * nrows + n] = p;               // (mu,alpha,beta,gamma)
    }
}

// ---------------------------------------------------------------------------
extern "C" void kernel_launch(void* const* d_in, const int* in_sizes, int n_in,
                              void* d_out, int out_size, void* d_ws, size_t ws_size,
                              hipStream_t stream) {
    const float* x     = (const float*)d_in[0];
    const float* coef0 = (const float*)d_in[1];
    const float* sb0   = (const float*)d_in[2];
    const float* ss0   = (const float*)d_in[3];
    const float* coef1 = (const float*)d_in[4];
    const float* sb1   = (const float*)d_in[5];
    const float* ss1   = (const float*)d_in[6];
    const float* coef2 = (const float*)d_in[7];
    const float* sb2   = (const float*)d_in[8];
    const float* ss2   = (const float*)d_in[9];
    float* out = (float*)d_out;

    const int nrows = in_sizes[0] / (512 * 512);   // 1024

    float* ws = (float*)d_ws;
    float* h0 = ws;                                // nrows*512
    float* W0 = h0 + (size_t)nrows * N_IN;         // 4608*64
    float* h1 = W0 + (size_t)K_TOT * N_HID;        // nrows*64

    // 1) mean over axis 1 (HBM-bound, 1.07 GB)
    mean_rows_kernel<<<nrows, 128, 0, stream>>>(x, h0);

    // 2) pack fused layer-0 weights (4608 x 64)
    {
        int tot = K_TOT * N_HID;
        pack_w0_kernel<<<(tot + 255) / 256, 256, 0, stream>>>(coef0, sb0, ss0, W0);
    }

    // 3) layer-0 GEMM via V_WMMA_F32_16X16X4_F32
    layer0_wmma_kernel<<<nrows / 16, 128, 0, stream>>>(h0, W0, h1);

    // 4) tail layers + softplus
    tail_kernel<<<(nrows + 255) / 256, 256, 0, stream>>>(
        h1, coef1, sb1, ss1, coef2, sb2, ss2, out, nrows);
}